// RecyclingEmbedder_28845000360676
// MI455X (gfx1250) — compile-verified
//
#include <hip/hip_runtime.h>
#include <hip/hip_bf16.h>

// RecyclingEmbedder binned-distance embedding for MI455X (gfx1250, wave32).
//
// out[i, j, :] = b + (one_hot(bin(|x_i - x_j|^2)) @ W^T)
//
// Store-bandwidth bound: 1.21 GB of f32 output @ 23.3 TB/s ~= 52 us floor.
// The K=15 one-hot projection is done bit-exactly with V_WMMA_F32_16X16X4_F32
// (A = one-hot rows in {0,1}, B = W^T slices, fp32 accumulate; exactly one
// nonzero product per row so no summation-order effects).

typedef __attribute__((ext_vector_type(2))) float v2f;
typedef __attribute__((ext_vector_type(8))) float v8f;

#define N_PTS    1536
#define D_PAIR   128
#define NUM_BINS 15
#define BIG_UP   100000000.0f          // reference's INF upper edge
#define JT       (N_PTS / 16)          // 96 j-tiles per row
#define NTILES   (N_PTS * JT)          // 147456 16-pair M-tiles
#define BLOCK    256
#define WPB      (BLOCK / 32)          // 8 waves per block (wave32)
#define NBLOCKS  4096                  // 32768 waves; ~4.5 tiles per wave

__global__ __launch_bounds__(BLOCK)
void recyc_embed_wmma(const float* __restrict__ x,
                      const float* __restrict__ W,     // [128, 15] row-major
                      const float* __restrict__ bias,  // [128]
                      float* __restrict__ out)         // [1536, 1536, 128]
{
    const int lane = threadIdx.x & 31;
    const int wid  = blockIdx.x * WPB + (threadIdx.x >> 5); // wave-uniform
    const int nw   = NBLOCKS * WPB;
    const int lm   = lane & 15;          // column / pair-in-half index
    const int hi   = lane >> 4;          // 0: lanes 0-15, 1: lanes 16-31

    // ---- squared bin edges; exact fp32 like the reference (edges are exact
    //      multiples of 0.25 so linspace matches bit-for-bit, squared in f32)
    float BSQ[NUM_BINS];
#pragma unroll
    for (int k = 0; k < NUM_BINS; ++k) {
        const float e = 3.25f + 1.25f * (float)k;
        BSQ[k] = e * e;
    }

    // ---- loop-invariant B fragments: B[k][n] = W[n][k], K padded 15->16 with 0.
    // ISA f32 A/B layout: value k = reg + 2*hi (within a K=4 chunk), n = lm.
    float Bx[8][4], By[8][4], bb[8];
#pragma unroll
    for (int nb = 0; nb < 8; ++nb) {
        const int n = nb * 16 + lm;
        bb[nb] = bias[n];
#pragma unroll
        for (int c = 0; c < 4; ++c) {
            const int k0 = 4 * c + 2 * hi;
            Bx[nb][c] = (k0     < NUM_BINS) ? W[n * NUM_BINS + k0]     : 0.0f;
            By[nb][c] = (k0 + 1 < NUM_BINS) ? W[n * NUM_BINS + k0 + 1] : 0.0f;
        }
    }

    for (int t = wid; t < NTILES; t += nw) {      // wave-uniform: EXEC stays full
        const int i  = t / JT;
        const int j0 = (t - i * JT) * 16;

        // pair (i, j0+lm): both lane halves compute the same d for their row
        const float xix = x[3 * i + 0], xiy = x[3 * i + 1], xiz = x[3 * i + 2];
        const int   j   = j0 + lm;
        const float dx  = xix - x[3 * j + 0];
        const float dy  = xiy - x[3 * j + 1];
        const float dz  = xiz - x[3 * j + 2];
        const float d   = dx * dx + dy * dy + dz * dz;

        // strict one-hot bin: BSQ[k] < d < BSQ[k+1] (last upper = 1e8)
        int idx = 255;                           // sentinel: no bin -> zero A row
#pragma unroll
        for (int k = 0; k < NUM_BINS; ++k) {
            const float up = (k + 1 < NUM_BINS) ? BSQ[k + 1] : BIG_UP;
            if (d > BSQ[k] && d < up) idx = k;
        }

        // A fragments (16x4 f32 per chunk): one-hot rows
        v2f A[4];
#pragma unroll
        for (int c = 0; c < 4; ++c) {
            const int k0 = 4 * c + 2 * hi;
            A[c].x = (idx == k0)     ? 1.0f : 0.0f;
            A[c].y = (idx == k0 + 1) ? 1.0f : 0.0f;
        }

        // C/D layout: VGPR r -> row (r + 8*hi), lane -> column lm (+ nb*16).
        // Single address VGPR; all 64 stores use immediate byte offsets.
        float* outp = out + ((size_t)(i * N_PTS + j0 + 8 * hi) * D_PAIR) + lm;

#pragma unroll
        for (int nb = 0; nb < 8; ++nb) {
            v8f acc = {};
#pragma unroll
            for (int c = 0; c < 4; ++c) {
                v2f Bf;
                Bf.x = Bx[nb][c];
                Bf.y = By[nb][c];
                acc = __builtin_amdgcn_wmma_f32_16x16x4_f32(
                    /*neg_a=*/false, A[c], /*neg_b=*/false, Bf,
                    /*c_mod=*/(short)0, acc, /*reuse_a=*/false, /*reuse_b=*/false);
            }
            const float bbn = bb[nb];
#pragma unroll
            for (int r = 0; r < 8; ++r)
                outp[(size_t)r * D_PAIR + nb * 16] = acc[r] + bbn;
        }
    }
}

extern "C" void kernel_launch(void* const* d_in, const int* in_sizes, int n_in,
                              void* d_out, int out_size, void* d_ws, size_t ws_size,
                              hipStream_t stream) {
    const float* x = (const float*)d_in[0];   // [1536, 3]
    const float* W = (const float*)d_in[1];   // [128, 15]
    const float* b = (const float*)d_in[2];   // [128]
    float* out = (float*)d_out;               // [1536, 1536, 128]
    recyc_embed_wmma<<<NBLOCKS, BLOCK, 0, stream>>>(x, W, b, out);
}